// FSQ_AdaptiveQuant_todo_fix_1271310319898
// MI455X (gfx1250) — compile-verified
//
#include <hip/hip_runtime.h>

// Problem constants (from reference)
#define CIN   768
#define CD    3
#define NC    15
#define NROWS (32 * 4096)          // B*T = 131072
#define NTILES (NROWS / 16)        // 8192 16-row tiles
#define NTN   (CIN / 16)           // 48 N-tiles of W_e

typedef float v2f __attribute__((ext_vector_type(2)));
typedef float v8f __attribute__((ext_vector_type(8)));

// ---------------------------------------------------------------------------
// K1: stream z once. zc = z @ W_c + b_c  (131072 x 3), plus global KDE
// histogram kde[3][15] += exp(-2*(center - zc)^2).
// One wave (32 lanes) per 16 rows: lane loads float4 chunks lane+32*j of the
// row (coalesced 512B per instruction), FMAs against W_c slices preloaded into
// registers, then a 5-step butterfly shfl_xor reduction gives every lane the
// full dot products.
// ---------------------------------------------------------------------------
__global__ __launch_bounds__(256) void k1_zc_kde(
    const float* __restrict__ z, const float* __restrict__ Wc,
    const float* __restrict__ bc, const float* __restrict__ cent,
    float* __restrict__ zc, float* __restrict__ kde)
{
    __shared__ float sHist[CD * NC];
    __shared__ float sCent[CD * NC];
    const int tid = threadIdx.x;
    if (tid < CD * NC) { sHist[tid] = 0.0f; sCent[tid] = cent[tid]; }
    __syncthreads();

    const int lane  = tid & 31;
    const int wave  = tid >> 5;
    const int gwave = blockIdx.x * 8 + wave;      // 1024 blocks * 8 waves = 8192

    // Per-lane W_c slices: element e = 4*(lane + 32*j) + t, j=0..5, t=0..3.
    // 72 floats in registers -> zero LDS traffic in the inner loop.
    float wc[6][4][3];
#pragma unroll
    for (int j = 0; j < 6; ++j)
#pragma unroll
        for (int t = 0; t < 4; ++t) {
            const int e = 4 * (lane + 32 * j) + t;
#pragma unroll
            for (int d = 0; d < 3; ++d) wc[j][t][d] = Wc[e * 3 + d];
        }
    const float bc0 = bc[0], bc1 = bc[1], bc2 = bc[2];

    const int r0 = gwave * 16;
    for (int rr = 0; rr < 16; ++rr) {
        const int r = r0 + rr;
        const float4* zr = (const float4*)(z + (size_t)r * CIN);
        float a0 = 0.f, a1 = 0.f, a2 = 0.f;
#pragma unroll
        for (int j = 0; j < 6; ++j) {
            const float4 v = zr[lane + 32 * j];
            a0 += v.x * wc[j][0][0] + v.y * wc[j][1][0] + v.z * wc[j][2][0] + v.w * wc[j][3][0];
            a1 += v.x * wc[j][0][1] + v.y * wc[j][1][1] + v.z * wc[j][2][1] + v.w * wc[j][3][1];
            a2 += v.x * wc[j][0][2] + v.y * wc[j][1][2] + v.z * wc[j][2][2] + v.w * wc[j][3][2];
        }
        // wave32 butterfly: every lane ends with the full row sums
#pragma unroll
        for (int m = 16; m >= 1; m >>= 1) {
            a0 += __shfl_xor(a0, m, 32);
            a1 += __shfl_xor(a1, m, 32);
            a2 += __shfl_xor(a2, m, 32);
        }
        a0 += bc0; a1 += bc1; a2 += bc2;
        if (lane == 0) {
            zc[r * 3 + 0] = a0; zc[r * 3 + 1] = a1; zc[r * 3 + 2] = a2;
        }
        // 45 KDE bins spread over 32 lanes (lanes 0..12 take a second bin)
        for (int i = lane; i < CD * NC; i += 32) {
            const int d = i / NC;
            const float zd = (d == 0) ? a0 : ((d == 1) ? a1 : a2);
            const float t = sCent[i] - zd;               // bandwidth 0.5 -> -0.5*(t/0.5)^2 = -2 t^2
            atomicAdd(&sHist[i], __expf(-2.0f * t * t)); // ds_add_f32
        }
    }
    __syncthreads();
    if (tid < CD * NC) atomicAdd(&kde[tid], sHist[tid]); // global_atomic_add_f32
}

// ---------------------------------------------------------------------------
// K2: scaled_centers = centers * kde / rowsum(kde); also writes the scalar
// second tuple element of the output.
// ---------------------------------------------------------------------------
__global__ void k2_scale(const float* __restrict__ kde, const float* __restrict__ cent,
                         float* __restrict__ sc, float* __restrict__ out_tail)
{
    __shared__ float ssum[CD];
    const int t = threadIdx.x;
    if (t < CD) {
        float s = 0.0f;
        for (int c = 0; c < NC; ++c) s += kde[t * NC + c];
        ssum[t] = s;
    }
    __syncthreads();
    if (t < CD * NC) sc[t] = cent[t] * kde[t] / ssum[t / NC];
    if (t == CD * NC) *out_tail = 0.0f;
}

// ---------------------------------------------------------------------------
// K3: quantize zc to nearest scaled center (first-min wins, like argmin), then
// z_q = q @ W_e + b_e via V_WMMA_F32_16X16X4_F32.
//   A (16x4, K=3 padded): lanes 0-15 hold {K0,K1} for M=lane, lanes 16-31 hold
//   {K2, 0} for M=lane-16 (ISA 32-bit A 16x4 layout).
//   B (4x16): lanes 0-15 hold {K0,K1} for N=lane, lanes 16-31 {K2, 0}.
//   C/D (16x16): VGPR v = row (v | hi*8), col = lane&15.
// B fragments are pre-packed (zero-padded) into LDS in exactly the per-lane
// WMMA layout, so the inner loop is one unconditional ds_load_b64 per tile —
// no EXEC-predicated LDS loads around the WMMA.
// One wave per 16-row tile, looping 48 N-tiles.
// ---------------------------------------------------------------------------
__global__ __launch_bounds__(256) void k3_quant_wmma(
    const float* __restrict__ zc, const float* __restrict__ sc,
    const float* __restrict__ We, const float* __restrict__ be,
    float* __restrict__ out)
{
    __shared__ v2f   sB[NTN * 32];    // 12 KB: packed B fragments [nt][lane]
    __shared__ float sBe[CIN];
    __shared__ float sSC[CD * NC];
    const int tid = threadIdx.x;

    // Pack zero-padded B fragments once per block (setup-only divergence).
    for (int i = tid; i < NTN * 32; i += 256) {
        const int nt = i >> 5;
        const int L  = i & 31;
        const int col = nt * 16 + (L & 15);
        v2f b;
        if (L < 16) { b[0] = We[0 * CIN + col]; b[1] = We[1 * CIN + col]; }
        else        { b[0] = We[2 * CIN + col]; b[1] = 0.0f;              }
        sB[i] = b;
    }
    for (int i = tid; i < CIN; i += 256) sBe[i] = be[i];
    if (tid < CD * NC)                   sSC[tid] = sc[tid];
    __syncthreads();

    const int  lane = tid & 31;
    const int  wave = tid >> 5;
    const int  tile = blockIdx.x * 8 + wave;   // 1024 blocks * 8 = 8192 tiles
    const int  r0   = tile * 16;
    const int  mrow = r0 + (lane & 15);
    const bool hi   = lane >= 16;

    // Quantize this row's 3 code dims (argmin over 15 scaled centers).
    float q[3];
#pragma unroll
    for (int d = 0; d < 3; ++d) {
        const float v = zc[mrow * 3 + d];
        float best = __builtin_inff();
        float bq = 0.0f;
#pragma unroll
        for (int c = 0; c < NC; ++c) {
            const float s  = sSC[d * NC + c];
            const float dd = __builtin_fabsf(v - s);
            if (dd < best) { best = dd; bq = s; }   // strict < == first-min (argmin)
        }
        q[d] = bq;
    }

    v2f A;
    A[0] = hi ? q[2] : q[0];
    A[1] = hi ? 0.0f : q[1];     // K=3 padded to 4 with zeros

    const int    cbase     = lane & 15;
    const size_t orow_base = (size_t)(r0 + (hi ? 8 : 0)) * CIN;

    for (int nt = 0; nt < NTN; ++nt) {          // 48 N-tiles
        const int col = nt * 16 + cbase;
        const v2f Bv  = sB[nt * 32 + lane];     // single ds_load_b64, uniform
        v8f C = {};
        C = __builtin_amdgcn_wmma_f32_16x16x4_f32(
                /*neg_a=*/false, A, /*neg_b=*/false, Bv,
                /*c_mod=*/(short)0, C, /*reuse_a=*/false, /*reuse_b=*/false);
        const float bias = sBe[col];
#pragma unroll
        for (int v = 0; v < 8; ++v)
            out[orow_base + (size_t)v * CIN + col] = C[v] + bias;
    }
}

// ---------------------------------------------------------------------------
extern "C" void kernel_launch(void* const* d_in, const int* in_sizes, int n_in,
                              void* d_out, int out_size, void* d_ws, size_t ws_size,
                              hipStream_t stream)
{
    (void)in_sizes; (void)n_in; (void)out_size; (void)ws_size;
    const float* z    = (const float*)d_in[0];
    const float* Wc   = (const float*)d_in[1];
    const float* bc   = (const float*)d_in[2];
    const float* We   = (const float*)d_in[3];
    const float* be   = (const float*)d_in[4];
    const float* cent = (const float*)d_in[5];
    float* out = (float*)d_out;

    // workspace layout: [0,64) kde bins, [64,128) scaled centers, [128,...) zc
    float* kde = (float*)d_ws;
    float* sc  = (float*)d_ws + 64;
    float* zc  = (float*)d_ws + 128;

    hipMemsetAsync(kde, 0, 64 * sizeof(float), stream);
    k1_zc_kde<<<NTILES / 8, 256, 0, stream>>>(z, Wc, bc, cent, zc, kde);
    k2_scale<<<1, 64, 0, stream>>>(kde, cent, sc, out + (size_t)NROWS * CIN);
    k3_quant_wmma<<<NTILES / 8, 256, 0, stream>>>(zc, sc, We, be, out);
}